// FromRGB_44135083934225
// MI455X (gfx1250) — compile-verified
//
#include <hip/hip_runtime.h>
#include <hip/hip_bf16.h>
#include <stddef.h>

typedef float v2f __attribute__((ext_vector_type(2)));
typedef float v4f __attribute__((ext_vector_type(4)));
typedef float v8f __attribute__((ext_vector_type(8)));

// ---------------------------------------------------------------------------
// Kernel 1: fused InverseHaar(up2) -> blur[1,3,3,1](down2) -> Haar(down2).
// Composed analytically into one stride-2 separable 4x4 filter bank:
//   new_in[b, 4k+c, m, n] = (1/256) * sum_s sum_{dj,di in 0..3}
//        Prow[frow(s)][rt(k)][dj] * Pcol[fcol(s)][ct(k)][di]
//        * x[b, c+4s, 2m-1+dj, 2n-1+di]        (zero-padded)
// ---------------------------------------------------------------------------
__global__ __launch_bounds__(256) void wavelet_composite_kernel(
    const float* __restrict__ x, float* __restrict__ new_in)
{
  const int n = threadIdx.x;            // output col 0..255
  const int m = blockIdx.x & 255;       // output row
  const int c = (blockIdx.x >> 8) & 3;  // channel within subband group
  const int b = blockIdx.x >> 10;       // batch

  // 1D composite taps (x16): Pt[f][t][dj]; f: 0 = low-pass 'l', 1 = high 'h'
  // t: 0 = output-low (Haar l), 1 = output-high (Haar h)
  const float Pt[2][2][4] = {
      {{1.f, 7.f, 7.f, 1.f},   {1.f, 5.f, -5.f, -1.f}},
      {{1.f, 1.f, -1.f, -1.f}, {1.f, -1.f, -1.f, 1.f}}};
  // subband s in {ll,lh,hl,hh}: row factor l,-h,l,h ; col factor l,l,-h,h
  const int   fri[4] = {0, 1, 0, 1};
  const float frs[4] = {1.f, -1.f, 1.f, 1.f};
  const int   fci[4] = {0, 0, 1, 1};
  const float fcs[4] = {1.f, 1.f, -1.f, 1.f};

  const int r0 = 2 * m - 1;
  const int c0 = 2 * n - 1;
  bool cok[4];
#pragma unroll
  for (int di = 0; di < 4; ++di) cok[di] = (unsigned)(c0 + di) < 512u;

  float acc0 = 0.f, acc1 = 0.f, acc2 = 0.f, acc3 = 0.f;

#pragma unroll
  for (int s = 0; s < 4; ++s) {
    const float* xs = x + (size_t)(b * 16 + c + 4 * s) * (512 * 512);
    float colL[4], colH[4];
#pragma unroll
    for (int dj = 0; dj < 4; ++dj) {
      const int  ri  = r0 + dj;
      const bool rok = (unsigned)ri < 512u;
      float sl = 0.f, sh = 0.f;
#pragma unroll
      for (int di = 0; di < 4; ++di) {
        const int ci = c0 + di;
        const float t = (rok && cok[di]) ? xs[(size_t)ri * 512 + ci] : 0.f;
        sl += Pt[fci[s]][0][di] * t;
        sh += Pt[fci[s]][1][di] * t;
      }
      colL[dj] = fcs[s] * sl;
      colH[dj] = fcs[s] * sh;
    }
#pragma unroll
    for (int dj = 0; dj < 4; ++dj) {
      const float rl = frs[s] * Pt[fri[s]][0][dj];
      const float rh = frs[s] * Pt[fri[s]][1][dj];
      acc0 += rl * colL[dj];  // k=0 (ll): row low,  col low
      acc1 += rh * colL[dj];  // k=1 (lh): row high, col low
      acc2 += rl * colH[dj];  // k=2 (hl): row low,  col high
      acc3 += rh * colH[dj];  // k=3 (hh): row high, col high
    }
  }
  const float sc = 1.f / 256.f;
  const size_t ob = ((size_t)(b * 16 + c) * 256 + m) * 256 + n;
  new_in[ob]                       = acc0 * sc;  // channel c
  new_in[ob + (size_t)4 * 65536]   = acc1 * sc;  // channel c+4
  new_in[ob + (size_t)8 * 65536]   = acc2 * sc;  // channel c+8
  new_in[ob + (size_t)12 * 65536]  = acc3 * sc;  // channel c+12
}

// ---------------------------------------------------------------------------
// Kernel 2: 1x1 conv (K=16) + bias + leaky-ReLU*sqrt2 via V_WMMA_F32_16X16X4_F32.
// Per wave: one 16-pixel tile (B resident in VGPRs, scale folded in), sweep
// all 32 out-channel tiles; 4 chained WMMAs cover K=16. f32 end to end.
// ---------------------------------------------------------------------------
__global__ __launch_bounds__(256) void conv1x1_wmma_kernel(
    const float* __restrict__ xin, const float* __restrict__ wgt,
    const float* __restrict__ bias, float* __restrict__ out)
{
  const int lane = threadIdx.x & 31;
  const int wave = blockIdx.x * (blockDim.x >> 5) + (threadIdx.x >> 5);
  // 32768 spatial tiles: [b:8][i:256][jt:16]
  const int jt = wave & 15;
  const int i  = (wave >> 4) & 255;
  const int b  = wave >> 12;
  const int n  = lane & 15;   // N column / A row
  const int hi = lane >> 4;   // half-wave select

  // B matrix (K=16 x N=16) as 4 chunks of 4x16; per ISA striping:
  // VGPR.x: lanes0-15 K=k0, lanes16-31 K=k0+2 ; VGPR.y: K=k0+1 / K=k0+3
  const float* xp = xin + (size_t)b * 16 * 65536 + (size_t)i * 256 + jt * 16 + n;
  v2f B[4];
#pragma unroll
  for (int k0 = 0; k0 < 4; ++k0) {
    const int cch = k0 * 4 + hi * 2;
    B[k0].x = 0.25f * xp[(size_t)cch * 65536];        // scale = 1/sqrt(16)
    B[k0].y = 0.25f * xp[(size_t)(cch + 1) * 65536];
  }

  const size_t obase0 =
      ((size_t)b * 512 + hi * 8) * 65536 + (size_t)i * 256 + jt * 16 + n;

  for (int ot = 0; ot < 32; ++ot) {
    v8f d = {0.f, 0.f, 0.f, 0.f, 0.f, 0.f, 0.f, 0.f};
#pragma unroll
    for (int k0 = 0; k0 < 4; ++k0) {
      const int kk = k0 * 4 + hi * 2;
      // A 16x4: lane M = n, K pair kk/kk+1 -> one aligned b64 load
      v2f a = *(const v2f*)(wgt + (size_t)(ot * 16 + n) * 16 + kk);
      d = __builtin_amdgcn_wmma_f32_16x16x4_f32(false, a, false, B[k0],
                                                (short)0, d, false, false);
    }
    // D VGPR v -> out channel ot*16 + hi*8 + v ; bias as two aligned b128s
    const v4f* bp = (const v4f*)(bias + ot * 16 + hi * 8);
    const v4f b0 = bp[0], b1 = bp[1];
    const float bb[8] = {b0.x, b0.y, b0.z, b0.w, b1.x, b1.y, b1.z, b1.w};
    const size_t ob = obase0 + (size_t)ot * 16 * 65536;
#pragma unroll
    for (int v = 0; v < 8; ++v) {
      float t = d[v] + bb[v];
      t = (t >= 0.f ? t : 0.2f * t) * 1.4142135623730951f;
      out[ob + (size_t)v * 65536] = t;
    }
  }
}

extern "C" void kernel_launch(void* const* d_in, const int* in_sizes, int n_in,
                              void* d_out, int out_size, void* d_ws, size_t ws_size,
                              hipStream_t stream) {
  (void)in_sizes; (void)n_in; (void)out_size; (void)d_ws; (void)ws_size;
  const float* x    = (const float*)d_in[0];  // [8,16,512,512]
  const float* w    = (const float*)d_in[1];  // [512,16,1,1]
  const float* bias = (const float*)d_in[2];  // [512]
  float* new_in = (float*)d_out;                            // [8,16,256,256]
  float* out    = new_in + (size_t)8 * 16 * 256 * 256;      // [8,512,256,256]

  // 8192 blocks = 8 batch * 4 c * 256 rows ; 256 threads = one output row
  wavelet_composite_kernel<<<8192, 256, 0, stream>>>(x, new_in);
  // 32768 waves (8 b * 256 rows * 16 col-tiles) / 8 waves per block
  conv1x1_wmma_kernel<<<4096, 256, 0, stream>>>(new_in, w, bias, out);
}